// RepresentationNetwork_49993419325458
// MI455X (gfx1250) — compile-verified
//
#include <hip/hip_runtime.h>
#include <math.h>

// Problem constants (from reference): B=2048, L=128, D=64, H=256, TOP_K=128
#define BN 2048
#define LN 128
#define DN 64
#define HN 256
#define KTOP 128

typedef __attribute__((ext_vector_type(2))) float v2f;
typedef __attribute__((ext_vector_type(8))) float v8f;

// ---------------------------------------------------------------------------
// f32 WMMA 16x16 tile MAC over kCount (multiple of 4) K-elements.
// A: row-major [16 x K] with leading dim lda (LDS)
// B: row-major [K x 16] with leading dim ldb (global or LDS)
// Layout per CDNA5 ISA 7.12.2 (16x4 f32 A, 4x16 f32 B, 16x16 f32 C/D).
// ---------------------------------------------------------------------------
__device__ __forceinline__ v8f wmma_mac(const float* __restrict__ A, int lda,
                                        const float* __restrict__ Bm, int ldb,
                                        int kCount, v8f acc) {
  const int lane = threadIdx.x & 31;
  const int half = lane >> 4;
  const int ln = lane & 15;
  for (int k0 = 0; k0 < kCount; k0 += 4) {
    const int ka = k0 + 2 * half;
    v2f a, b;
    a[0] = A[ln * lda + ka];
    a[1] = A[ln * lda + ka + 1];
    b[0] = Bm[ka * ldb + ln];
    b[1] = Bm[(ka + 1) * ldb + ln];
    acc = __builtin_amdgcn_wmma_f32_16x16x4_f32(false, a, false, b,
                                                (short)0, acc, false, false);
  }
  return acc;
}

// Same but B given transposed: B element (k,n) = Bt[n*ldb + k]  (for S = Y @ x^T)
__device__ __forceinline__ v8f wmma_mac_bt(const float* __restrict__ A, int lda,
                                           const float* __restrict__ Bt, int ldb,
                                           int kCount, v8f acc) {
  const int lane = threadIdx.x & 31;
  const int half = lane >> 4;
  const int ln = lane & 15;
  for (int k0 = 0; k0 < kCount; k0 += 4) {
    const int ka = k0 + 2 * half;
    v2f a, b;
    a[0] = A[ln * lda + ka];
    a[1] = A[ln * lda + ka + 1];
    b[0] = Bt[ln * ldb + ka];
    b[1] = Bt[ln * ldb + ka + 1];
    acc = __builtin_amdgcn_wmma_f32_16x16x4_f32(false, a, false, b,
                                                (short)0, acc, false, false);
  }
  return acc;
}

__device__ __forceinline__ unsigned f2key(float f) {
  unsigned u = __float_as_uint(f);
  return (u & 0x80000000u) ? ~u : (u | 0x80000000u);
}

// ---------------------------------------------------------------------------
// Kernel A: precompute M = Wq @ Wk^T  [64x64], u = Wq@bk, v = Wk@bq, c = bq.bk
// ---------------------------------------------------------------------------
__global__ __launch_bounds__(256) void repnet_pre(
    const float* __restrict__ Wq, const float* __restrict__ bq,
    const float* __restrict__ Wk, const float* __restrict__ bk,
    float* __restrict__ Mm, float* __restrict__ uv, float* __restrict__ vv,
    float* __restrict__ cv) {
  int t = blockIdx.x * 256 + threadIdx.x;  // 0..4095
  int d = t >> 6, e = t & 63;
  float acc = 0.f;
  for (int h = 0; h < HN; ++h) acc += Wq[d * HN + h] * Wk[e * HN + h];
  Mm[t] = acc;
  if (e == 0) {
    float su = 0.f, sv = 0.f;
    for (int h = 0; h < HN; ++h) {
      su += Wq[d * HN + h] * bk[h];
      sv += Wk[d * HN + h] * bq[h];
    }
    uv[d] = su;
    vv[d] = sv;
  }
  if (t == 0) {
    float c = 0.f;
    for (int h = 0; h < HN; ++h) c += bq[h] * bk[h];
    cv[0] = c;
  }
}

// ---------------------------------------------------------------------------
// Kernel B: one workgroup (256 threads = 8 wave32) per batch element.
// ---------------------------------------------------------------------------
__global__ __launch_bounds__(256) void repnet_main(
    const float* __restrict__ x, const float* __restrict__ Mm,
    const float* __restrict__ uv, const float* __restrict__ vv,
    const float* __restrict__ cv, const float* __restrict__ phi_W1,
    const float* __restrict__ phi_b1, const float* __restrict__ phi_W2,
    const float* __restrict__ phi_b2, const float* __restrict__ xi_W1,
    const float* __restrict__ xi_b1, const float* __restrict__ xi_W2,
    const float* __restrict__ xi_b2, const float* __restrict__ rho_W1,
    const float* __restrict__ rho_b1, const float* __restrict__ rho_W2,
    const float* __restrict__ rho_b2, float* __restrict__ out) {
  extern __shared__ float sm[];
  float* XB = sm;                         // [128][64]  8192
  float* Yb = sm + 8192;                  // [128][64]  8192
  float* Sb = sm + 16384;                 // [128][128] 16384 (reused as APAIR)
  float* XU = sm + 32768;                 // 128
  float* XV = sm + 32896;                 // 128
  float* MK = sm + 33024;                 // 128 (token mask)
  float* WSEL = sm + 33152;               // 128 topk vals -> softmax weights
  float* WSf = sm + 33280;                // 128 self-weights
  float* WPf = sm + 33408;                // 128 pair-weights
  int* IDX = (int*)(sm + 33536);          // 128 selected flat indices
  float* HSELF = sm + 33664;              // 256 pooled hidden (self)
  float* HPAIR = sm + 33920;              // 256 pooled hidden (pair)
  float* POOL = sm + 34176;               // 256 pooled output
  unsigned* HIST = (unsigned*)(sm + 34432);  // 256 radix bins
  float* CT = sm + 34688;                 // 16 counters/comm
  unsigned* CTu = (unsigned*)CT;

  const int t = threadIdx.x;
  const int b = blockIdx.x;
  const int wave = t >> 5;
  const int lane = t & 31;
  const int half = lane >> 4;
  const int ln = lane & 15;

  // hint the shared weight matrices into cache early (global_prefetch_b8)
  if (t == 0) {
    __builtin_prefetch(phi_W1, 0, 1);
    __builtin_prefetch(xi_W1, 0, 1);
    __builtin_prefetch(phi_W2, 0, 1);
    __builtin_prefetch(xi_W2, 0, 1);
  }

  // ---- B1: load x_b into LDS (float4) + zero accumulators -----------------
  {
    const float4* xsrc = (const float4*)(x + (size_t)b * LN * DN);
    float4* xdst = (float4*)XB;
    for (int i = t; i < LN * DN / 4; i += 256) xdst[i] = xsrc[i];
  }
  HSELF[t] = 0.f;
  HPAIR[t] = 0.f;
  if (t < 16) CT[t] = 0.f;
  __syncthreads();

  // ---- token mask + per-row bias dots ------------------------------------
  if (t < LN) {
    float s = 0.f, du = 0.f, dv = 0.f;
    for (int d = 0; d < DN; ++d) {
      float xd = XB[t * DN + d];
      s += fabsf(xd);
      du += xd * uv[d];
      dv += xd * vv[d];
    }
    MK[t] = (s != 0.f) ? 1.f : 0.f;
    XU[t] = du;
    XV[t] = dv;
  }
  __syncthreads();

  // ---- B2: Y = XB @ M  [128x64] : 32 tiles, 8 waves -----------------------
  for (int tile = wave; tile < 32; tile += 8) {
    int mt = tile >> 2, nt = tile & 3;
    v8f acc = {};
    acc = wmma_mac(XB + mt * 16 * DN, DN, Mm + nt * 16, DN, DN, acc);
    for (int r = 0; r < 8; ++r)
      Yb[(mt * 16 + r + 8 * half) * DN + nt * 16 + ln] = acc[r];
  }
  __syncthreads();

  // ---- B3: S = scale*(Y @ XB^T + xu + xv + c), masked ---------------------
  {
    const float cc = cv[0];
    const float scale = 0.0625f;  // 256^-0.5
    for (int tile = wave; tile < 64; tile += 8) {
      int mt = tile >> 3, nt = tile & 7;
      v8f acc = {};
      acc = wmma_mac_bt(Yb + mt * 16 * DN, DN, XB + nt * 16 * DN, DN, DN, acc);
      for (int r = 0; r < 8; ++r) {
        int row = mt * 16 + r + 8 * half;
        int col = nt * 16 + ln;
        float v = scale * (acc[r] + XU[row] + XV[col] + cc);
        if (MK[row] * MK[col] == 0.f) v = -__builtin_inff();
        Sb[row * LN + col] = v;
      }
    }
  }
  __syncthreads();

  // ---- B4: top-128 via 4-pass MSB radix select ----------------------------
  unsigned prefKey = 0, prefMask = 0;
  int need = KTOP;
  for (int pass = 0; pass < 4; ++pass) {
    const int shift = 24 - 8 * pass;
    for (int i = t; i < 256; i += 256) HIST[i] = 0u;
    __syncthreads();
    for (int i = t; i < LN * LN; i += 256) {
      unsigned key = f2key(Sb[i]);
      if ((key & prefMask) == prefKey)
        atomicAdd(&HIST[(key >> shift) & 255u], 1u);
    }
    __syncthreads();
    if (t == 0) {
      int cum = 0, bsel = 0;
      for (int bb = 255; bb >= 0; --bb) {
        int h = (int)HIST[bb];
        if (cum + h >= need) { bsel = bb; break; }
        cum += h;
      }
      CTu[0] = (unsigned)bsel;
      CTu[1] = (unsigned)(need - cum);
    }
    __syncthreads();
    unsigned bsel = CTu[0];
    need = (int)CTu[1];
    prefKey |= bsel << shift;
    prefMask |= 0xFFu << shift;
    __syncthreads();
  }
  // collect: strictly greater first, then ties up to 'need'
  if (t == 0) { CTu[2] = 0u; CTu[3] = 0u; }
  __syncthreads();
  for (int i = t; i < LN * LN; i += 256) {
    unsigned key = f2key(Sb[i]);
    if (key > prefKey) {
      unsigned slot = atomicAdd(&CTu[2], 1u);
      IDX[slot] = i;
      WSEL[slot] = Sb[i];
    }
  }
  __syncthreads();
  for (int i = t; i < LN * LN; i += 256) {
    unsigned key = f2key(Sb[i]);
    if (key == prefKey) {
      unsigned e = atomicAdd(&CTu[3], 1u);
      if ((int)e < need) {
        unsigned slot = atomicAdd(&CTu[2], 1u);
        IDX[slot] = i;
        WSEL[slot] = Sb[i];
      }
    }
  }
  __syncthreads();

  // ---- B5: softmax over the 128 selected + self/pair split ----------------
  if (t == 0) {
    float mx = -__builtin_inff();
    for (int i = 0; i < KTOP; ++i) mx = fmaxf(mx, WSEL[i]);
    CT[4] = mx;
  }
  __syncthreads();
  if (t < KTOP) WSEL[t] = expf(WSEL[t] - CT[4]);
  __syncthreads();
  if (t == 0) {
    float s = 0.f;
    for (int i = 0; i < KTOP; ++i) s += WSEL[i];
    CT[5] = s;
  }
  __syncthreads();
  if (t < KTOP) {
    float w = WSEL[t] / CT[5];
    WSEL[t] = w;
    int idx = IDX[t];
    int row = idx >> 7, col = idx & 127;
    bool self = (row == col);
    WSf[t] = self ? w : 0.f;
    WPf[t] = self ? 0.f : w;
  }
  __syncthreads();
  if (t == 0) {
    float a = 0.f, p = 0.f;
    for (int i = 0; i < KTOP; ++i) { a += WSf[i]; p += WPf[i]; }
    CT[6] = a;
    CT[7] = p;
  }
  __syncthreads();

  // ---- B6: build APAIR = [x_i | x_j]  (reuse Sb region) -------------------
  float* APAIR = Sb;
  {
    int p = t >> 1;   // pair 0..127
    int hh = t & 1;   // 0: x_i, 1: x_j
    int idx = IDX[p];
    int src = hh ? (idx & 127) : (idx >> 7);
    const float4* s4 = (const float4*)(XB + src * DN);
    float4* d4 = (float4*)(APAIR + p * 128 + hh * 64);
    for (int d = 0; d < DN / 4; ++d) d4[d] = s4[d];
  }
  __syncthreads();

  // ---- B7: weighted-pooled ReLU hiddens via WMMA + ds_add_f32 -------------
  // self path: A = x_i (cols 0..63), B = phi_W1 [64,256]
  for (int tile = wave; tile < 128; tile += 8) {
    int mt = tile >> 4, nt = tile & 15;
    v8f acc = {};
    acc = wmma_mac(APAIR + mt * 16 * 128, 128, phi_W1 + nt * 16, HN, DN, acc);
    float bias = phi_b1[nt * 16 + ln];
    float part = 0.f;
    for (int r = 0; r < 8; ++r) {
      int m = mt * 16 + r + 8 * half;
      part += WSf[m] * fmaxf(acc[r] + bias, 0.f);
    }
    atomicAdd(&HSELF[nt * 16 + ln], part);
  }
  // pair path: A = [x_i|x_j] (128 cols), B = xi_W1 [128,256]
  for (int tile = wave; tile < 128; tile += 8) {
    int mt = tile >> 4, nt = tile & 15;
    v8f acc = {};
    acc = wmma_mac(APAIR + mt * 16 * 128, 128, xi_W1 + nt * 16, HN, 128, acc);
    float bias = xi_b1[nt * 16 + ln];
    float part = 0.f;
    for (int r = 0; r < 8; ++r) {
      int m = mt * 16 + r + 8 * half;
      part += WPf[m] * fmaxf(acc[r] + bias, 0.f);
    }
    atomicAdd(&HPAIR[nt * 16 + ln], part);
  }
  __syncthreads();

  // ---- B8: pooled = HSELF@phi_W2 + HPAIR@xi_W2 + weighted biases ----------
  {
    float acc = CT[6] * phi_b2[t] + CT[7] * xi_b2[t];
    for (int i = 0; i < HN; ++i)
      acc += HSELF[i] * phi_W2[i * HN + t] + HPAIR[i] * xi_W2[i * HN + t];
    POOL[t] = acc;
  }
  __syncthreads();

  // ---- B9: rho MLP (two 256x256 matvecs) ----------------------------------
  {
    float a = rho_b1[t];
    for (int i = 0; i < HN; ++i) a += POOL[i] * rho_W1[i * HN + t];
    HSELF[t] = fmaxf(a, 0.f);  // reuse as rho hidden
  }
  __syncthreads();
  {
    float a = rho_b2[t];
    for (int i = 0; i < HN; ++i) a += HSELF[i] * rho_W2[i * HN + t];
    out[(size_t)b * HN + t] = a;
  }
}

// ---------------------------------------------------------------------------
extern "C" void kernel_launch(void* const* d_in, const int* in_sizes, int n_in,
                              void* d_out, int out_size, void* d_ws,
                              size_t ws_size, hipStream_t stream) {
  const float* x      = (const float*)d_in[0];
  const float* Wq     = (const float*)d_in[1];
  const float* bq     = (const float*)d_in[2];
  const float* Wk     = (const float*)d_in[3];
  const float* bk     = (const float*)d_in[4];
  const float* phi_W1 = (const float*)d_in[5];
  const float* phi_b1 = (const float*)d_in[6];
  const float* phi_W2 = (const float*)d_in[7];
  const float* phi_b2 = (const float*)d_in[8];
  const float* xi_W1  = (const float*)d_in[9];
  const float* xi_b1  = (const float*)d_in[10];
  const float* xi_W2  = (const float*)d_in[11];
  const float* xi_b2  = (const float*)d_in[12];
  const float* rho_W1 = (const float*)d_in[13];
  const float* rho_b1 = (const float*)d_in[14];
  const float* rho_W2 = (const float*)d_in[15];
  const float* rho_b2 = (const float*)d_in[16];

  float* Mm = (float*)d_ws;      // 64*64
  float* uv = Mm + 64 * 64;      // 64
  float* vv = uv + 64;           // 64
  float* cv = vv + 64;           // 1

  repnet_pre<<<16, 256, 0, stream>>>(Wq, bq, Wk, bk, Mm, uv, vv, cv);

  const size_t smemBytes = 34704 * sizeof(float);  // ~139 KB dynamic LDS
  repnet_main<<<BN, 256, smemBytes, stream>>>(
      x, Mm, uv, vv, cv, phi_W1, phi_b1, phi_W2, phi_b2, xi_W1, xi_b1, xi_W2,
      xi_b2, rho_W1, rho_b1, rho_W2, rho_b2, (float*)d_out);
}